// RandTransformer2DModel_Pattern_0_1_2_79534204387667
// MI455X (gfx1250) — compile-verified
//
#include <hip/hip_runtime.h>
#include <hip/hip_bf16.h>
#include <stdint.h>

#define NUM_BLOCKS 64
#define RAND_RATIO 0.5f
#define RATIO 0.7f

typedef __attribute__((ext_vector_type(4))) float v4f;

// ---------------------------------------------------------------------------
// JAX Threefry-2x32 (exact rotation/key schedule used by jax.random)
// ---------------------------------------------------------------------------
__device__ __forceinline__ uint32_t rotl32(uint32_t x, int r) {
  return (x << r) | (x >> (32 - r));
}

__device__ __forceinline__ void tf2x32(uint32_t k0, uint32_t k1,
                                       uint32_t x0, uint32_t x1,
                                       uint32_t& o0, uint32_t& o1) {
  const uint32_t ks0 = k0, ks1 = k1, ks2 = k0 ^ k1 ^ 0x1BD11BDAu;
#define TF_ROUND(r) { x0 += x1; x1 = rotl32(x1, r); x1 ^= x0; }
  x0 += ks0; x1 += ks1;
  TF_ROUND(13) TF_ROUND(15) TF_ROUND(26) TF_ROUND(6)
  x0 += ks1; x1 += ks2 + 1u;
  TF_ROUND(17) TF_ROUND(29) TF_ROUND(16) TF_ROUND(24)
  x0 += ks2; x1 += ks0 + 2u;
  TF_ROUND(13) TF_ROUND(15) TF_ROUND(26) TF_ROUND(6)
  x0 += ks0; x1 += ks1 + 3u;
  TF_ROUND(17) TF_ROUND(29) TF_ROUND(16) TF_ROUND(24)
  x0 += ks1; x1 += ks2 + 4u;
  TF_ROUND(13) TF_ROUND(15) TF_ROUND(26) TF_ROUND(6)
  x0 += ks2; x1 += ks0 + 5u;
#undef TF_ROUND
  o0 = x0; o1 = x1;
}

// ---------------------------------------------------------------------------
// Branch-free ErfInv32 (XLA's polynomial, per-lane coefficient select:
// compiles to v_cndmask + single Horner FMA chain, no divergence).
// ---------------------------------------------------------------------------
__device__ __forceinline__ float jax_erfinv(float x) {
  float w = -log1pf(-x * x);
  const bool t = (w < 5.0f);
  float wa = t ? (w - 2.5f) : (__fsqrt_rn(w) - 3.0f);
  float p;
  p =               t ?  2.81022636e-08f : -0.000200214257f;
  p = fmaf(p, wa,   t ?  3.43273939e-07f :  0.000100950558f);
  p = fmaf(p, wa,   t ? -3.5233877e-06f  :  0.00134934322f);
  p = fmaf(p, wa,   t ? -4.39150654e-06f : -0.00367342844f);
  p = fmaf(p, wa,   t ?  0.00021858087f  :  0.00573950773f);
  p = fmaf(p, wa,   t ? -0.00125372503f  : -0.0076224613f);
  p = fmaf(p, wa,   t ? -0.00417768164f  :  0.00943887047f);
  p = fmaf(p, wa,   t ?  0.246640727f    :  1.00167406f);
  p = fmaf(p, wa,   t ?  1.50140941f     :  2.83297682f);
  return p * x;
}

// bits -> uniform exactly like jax (mantissa trick), then to N(0,1)
// via jax.random.normal's  sqrt(2)*erfinv(uniform(-1+ulp, 1)).
__device__ __forceinline__ float bits_to_normal(uint32_t b) {
  float u01 = __uint_as_float((b >> 9) | 0x3f800000u) - 1.0f;   // [0,1)
  const float lo = -0.99999994f;                                 // nextafter(-1,0)
  float u = fmaf(u01, 1.0f - lo, lo);
  u = fmaxf(u, lo);
  return 1.41421356237f * jax_erfinv(u);
}

// ---------------------------------------------------------------------------
// Kernel A: replay the 64 scalar branch decisions on-device; find last-true
// block j*, stash {found, scale = 0.7^(63-j*) or 0.7^64, kh(j*), ke(j*)}.
// ---------------------------------------------------------------------------
__global__ void prng_setup_kernel(uint32_t* __restrict__ ws) {
  __shared__ uint32_t s_branch[NUM_BLOCKS];
  __shared__ uint32_t s_kh0[NUM_BLOCKS], s_kh1[NUM_BLOCKS];
  __shared__ uint32_t s_ke0[NUM_BLOCKS], s_ke1[NUM_BLOCKS];

  const int i = threadIdx.x;               // 64 threads
  const uint32_t B0 = 0u, B1 = 42u;        // jax.random.key(42) => (0, 42)

  // keys = split(base, 64): count = iota(128), halves pair (j, 64+j).
  // bits = concat(out0[0:64], out1[0:64]); key_i = (bits[2i], bits[2i+1]).
  uint32_t k0, k1;
  if (i < 32) {
    uint32_t a0, d0, a1, d1;
    tf2x32(B0, B1, (uint32_t)(2 * i),     64u + (uint32_t)(2 * i),     a0, d0);
    tf2x32(B0, B1, (uint32_t)(2 * i + 1), 64u + (uint32_t)(2 * i + 1), a1, d1);
    k0 = a0; k1 = a1;                      // both from first-output half
  } else {
    uint32_t j = (uint32_t)(2 * i) - 64u;  // indices into second-output half
    uint32_t a, b, a2, b2;
    tf2x32(B0, B1, j,      64u + j,      a,  b);
    tf2x32(B0, B1, j + 1u, 64u + j + 1u, a2, b2);
    k0 = b; k1 = b2;
  }

  // split(key_i, 3): count = iota(6), pairs (j, 3+j), j=0..2.
  // bits = [c0,c1,c2,d0,d1,d2]; kb=(c0,c1), kh=(c2,d0), ke=(d1,d2).
  uint32_t c0, d0, c1, d1, c2, d2;
  tf2x32(k0, k1, 0u, 3u, c0, d0);
  tf2x32(k0, k1, 1u, 4u, c1, d1);
  tf2x32(k0, k1, 2u, 5u, c2, d2);

  // uniform(kb, ()): odd count padded -> threefry(kb, (0,0)), take word 0.
  uint32_t u0, u1;
  tf2x32(c0, c1, 0u, 0u, u0, u1);
  float u = __uint_as_float((u0 >> 9) | 0x3f800000u) - 1.0f;

  s_branch[i] = (u < RAND_RATIO) ? 1u : 0u;
  s_kh0[i] = c2; s_kh1[i] = d0;
  s_ke0[i] = d1; s_ke1[i] = d2;
  __syncthreads();

  if (i == 0) {
    int jstar = -1;
    for (int t = NUM_BLOCKS - 1; t >= 0; --t)
      if (s_branch[t]) { jstar = t; break; }
    float scale = (jstar >= 0) ? powf(RATIO, (float)(NUM_BLOCKS - 1 - jstar))
                               : powf(RATIO, (float)NUM_BLOCKS);
    ws[0] = (jstar >= 0) ? 1u : 0u;
    ws[1] = __float_as_uint(scale);
    ws[2] = (jstar >= 0) ? s_kh0[jstar] : 0u;
    ws[3] = (jstar >= 0) ? s_kh1[jstar] : 0u;
    ws[4] = (jstar >= 0) ? s_ke0[jstar] : 0u;
    ws[5] = (jstar >= 0) ? s_ke1[jstar] : 0u;
  }
}

// ---------------------------------------------------------------------------
// Kernel B: out[i] (i < N) with N = 2*Npair.
//  hot path  : out[i] = s*(Nh[i] + Ne[i])  -- write-only, counter-based PRNG.
//              jax pairing: threefry(k, (i, Npair+i)) yields bits for element
//              i (word0) and element i+Npair (word1) -> one hash, two outputs.
//              Streaming output written with non-temporal b128 stores.
//  cold path : out[i] = 0.7^64*(h[i]+e[i]) staged through LDS with CDNA5
//              global_load_async_to_lds_b128 + s_wait_asynccnt.
// ---------------------------------------------------------------------------
#define TPB 256

__global__ __launch_bounds__(TPB) void rand_blend_kernel(
    const float* __restrict__ hin, const float* __restrict__ ein,
    float* __restrict__ out, const uint32_t* __restrict__ ws, int Npair) {
  __shared__ float s_h[2 * TPB * 4];   // [lo 1024 | hi 1024] floats
  __shared__ float s_e[2 * TPB * 4];

  const uint32_t found = ws[0];
  const float scale = __uint_as_float(ws[1]);
  const int i0 = (int)(blockIdx.x * TPB + threadIdx.x) * 4;

  if (i0 + 3 < Npair) {
    if (found) {
      const uint32_t kh0 = ws[2], kh1 = ws[3], ke0 = ws[4], ke1 = ws[5];
      v4f lo, hi;
#pragma unroll
      for (int j = 0; j < 4; ++j) {
        const uint32_t i = (uint32_t)(i0 + j);
        uint32_t bh0, bh1, be0, be1;
        tf2x32(kh0, kh1, i, (uint32_t)Npair + i, bh0, bh1);
        tf2x32(ke0, ke1, i, (uint32_t)Npair + i, be0, be1);
        lo[j] = scale * (bits_to_normal(bh0) + bits_to_normal(be0));
        hi[j] = scale * (bits_to_normal(bh1) + bits_to_normal(be1));
      }
      __builtin_nontemporal_store(lo, (v4f*)(out + i0));
      __builtin_nontemporal_store(hi, (v4f*)(out + Npair + i0));
    } else {
      // --- CDNA5 async copy: global -> LDS, ASYNCcnt-tracked ---
      const int t4 = (int)threadIdx.x * 4;
      // low 32 bits of a generic LDS pointer == LDS byte address
      uint32_t a_hlo = (uint32_t)(uintptr_t)&s_h[t4];
      uint32_t a_hhi = (uint32_t)(uintptr_t)&s_h[TPB * 4 + t4];
      uint32_t a_elo = (uint32_t)(uintptr_t)&s_e[t4];
      uint32_t a_ehi = (uint32_t)(uintptr_t)&s_e[TPB * 4 + t4];
      const float* g_hlo = hin + i0;
      const float* g_hhi = hin + Npair + i0;
      const float* g_elo = ein + i0;
      const float* g_ehi = ein + Npair + i0;
      asm volatile("global_load_async_to_lds_b128 %0, %1, off"
                   :: "v"(a_hlo), "v"(g_hlo) : "memory");
      asm volatile("global_load_async_to_lds_b128 %0, %1, off"
                   :: "v"(a_hhi), "v"(g_hhi) : "memory");
      asm volatile("global_load_async_to_lds_b128 %0, %1, off"
                   :: "v"(a_elo), "v"(g_elo) : "memory");
      asm volatile("global_load_async_to_lds_b128 %0, %1, off"
                   :: "v"(a_ehi), "v"(g_ehi) : "memory");
      asm volatile("s_wait_asynccnt 0x0" ::: "memory");

      v4f hl = *(v4f*)&s_h[t4];
      v4f hh = *(v4f*)&s_h[TPB * 4 + t4];
      v4f el = *(v4f*)&s_e[t4];
      v4f eh = *(v4f*)&s_e[TPB * 4 + t4];
      v4f olo = scale * (hl + el);
      v4f ohi = scale * (hh + eh);
      __builtin_nontemporal_store(olo, (v4f*)(out + i0));
      __builtin_nontemporal_store(ohi, (v4f*)(out + Npair + i0));
    }
  } else {
    // scalar tail (also covers non-multiple-of-4 sizes)
    for (int j = 0; j < 4; ++j) {
      const int i = i0 + j;
      if (i >= Npair) break;
      if (found) {
        const uint32_t kh0 = ws[2], kh1 = ws[3], ke0 = ws[4], ke1 = ws[5];
        uint32_t bh0, bh1, be0, be1;
        tf2x32(kh0, kh1, (uint32_t)i, (uint32_t)(Npair + i), bh0, bh1);
        tf2x32(ke0, ke1, (uint32_t)i, (uint32_t)(Npair + i), be0, be1);
        out[i]         = scale * (bits_to_normal(bh0) + bits_to_normal(be0));
        out[Npair + i] = scale * (bits_to_normal(bh1) + bits_to_normal(be1));
      } else {
        out[i]         = scale * (hin[i] + ein[i]);
        out[Npair + i] = scale * (hin[Npair + i] + ein[Npair + i]);
      }
    }
  }
}

extern "C" void kernel_launch(void* const* d_in, const int* in_sizes, int n_in,
                              void* d_out, int out_size, void* d_ws, size_t ws_size,
                              hipStream_t stream) {
  const float* h = (const float*)d_in[0];
  const float* e = (const float*)d_in[1];
  float* out = (float*)d_out;
  uint32_t* ws = (uint32_t*)d_ws;

  const int Npair = out_size / 2;                  // 12,582,912 for 2x4096x3072
  prng_setup_kernel<<<1, NUM_BLOCKS, 0, stream>>>(ws);
  const int pairs_per_block = TPB * 4;
  const int nblk = (Npair + pairs_per_block - 1) / pairs_per_block;
  rand_blend_kernel<<<nblk, TPB, 0, stream>>>(h, e, out, ws, Npair);
}